// GraphNeuralOperatorBlock_15006615732852
// MI455X (gfx1250) — compile-verified
//
#include <hip/hip_runtime.h>
#include <hip/hip_bf16.h>
#include <cstdint>

typedef __attribute__((ext_vector_type(2))) float v2f;
typedef __attribute__((ext_vector_type(8))) float v8f;

#define NN 4096
#define CC 256
#define HH 8
#define DH 32
#define MASK_WPR 128   // N/32 words per row

__device__ __forceinline__ v8f wmma_f32(v2f a, v2f b, v8f c) {
    return __builtin_amdgcn_wmma_f32_16x16x4_f32(false, a, false, b, (short)0, c, false, false);
}

// ---------------- mask build ----------------
__global__ void zero_u32_kernel(uint32_t* p, int n) {
    int i = blockIdx.x * blockDim.x + threadIdx.x;
    if (i < n) p[i] = 0u;
}

__global__ void scatter_edges_kernel(const int* __restrict__ ei, int E, uint32_t* __restrict__ mask) {
    int e = blockIdx.x * blockDim.x + threadIdx.x;
    if (e >= E) return;
    int s = ei[e];
    int d = ei[E + e];
    atomicOr(&mask[s * MASK_WPR + (d >> 5)], 1u << (d & 31));
}

// ---------------- layer norm (1 block = 1 row, 256 threads) ----------------
__global__ void ln_kernel(const float* __restrict__ x, const float* __restrict__ g,
                          const float* __restrict__ b, float* __restrict__ out) {
    int row = blockIdx.x, t = threadIdx.x;
    float v = x[row * CC + t];
    __shared__ float red[CC];
    red[t] = v;
    __syncthreads();
    #pragma unroll
    for (int s = 128; s > 0; s >>= 1) {
        if (t < s) red[t] += red[t + s];
        __syncthreads();
    }
    float mean = red[0] * (1.0f / CC);
    __syncthreads();
    float d = v - mean;
    red[t] = d * d;
    __syncthreads();
    #pragma unroll
    for (int s = 128; s > 0; s >>= 1) {
        if (t < s) red[t] += red[t + s];
        __syncthreads();
    }
    float var = red[0] * (1.0f / CC);
    out[row * CC + t] = d * rsqrtf(var + 1e-5f) * g[t] + b[t];
}

// ---------------- generic WMMA fp32 GEMM: out[M,Nout] = A[M,K] @ W[Nout,K]^T ----------------
// epi: 0 = +bias, 1 = +bias+residual, 2 = +bias then exact GELU
__global__ void gemm_wmma_kernel(const float* __restrict__ A, const float* __restrict__ W,
                                 const float* __restrict__ bias, const float* __restrict__ resid,
                                 float* __restrict__ out, int M, int Nout, int K, int epi) {
    int lane = threadIdx.x & 31;
    int wid  = threadIdx.x >> 5;
    int ntn  = Nout >> 4;
    int tile = blockIdx.x * (blockDim.x >> 5) + wid;
    if (tile >= (M >> 4) * ntn) return;   // wave-uniform exit: EXEC stays all-ones for WMMA
    int m0 = (tile / ntn) << 4;
    int n0 = (tile % ntn) << 4;

    int mr = lane & 15;          // A row within tile / W row within tile
    int kb = (lane >> 4) << 1;   // K sub-offset per half-wave (ISA fp32 A/B layout)
    const float* arow = A + (size_t)(m0 + mr) * K + kb;
    const float* wrow = W + (size_t)(n0 + mr) * K + kb;

    v8f c = {};
    for (int k0 = 0; k0 < K; k0 += 4) {
        v2f a; a.x = arow[k0]; a.y = arow[k0 + 1];
        v2f b; b.x = wrow[k0]; b.y = wrow[k0 + 1];
        c = wmma_f32(a, b, c);
    }

    int col = n0 + (lane & 15);
    float bv = bias[col];
    #pragma unroll
    for (int r = 0; r < 8; ++r) {
        int row = m0 + r + ((lane >> 4) << 3);
        float v = c[r] + bv;
        if (epi == 1) v += resid[(size_t)row * Nout + col];
        else if (epi == 2) v = 0.5f * v * (1.0f + erff(v * 0.70710678118654752f));
        out[(size_t)row * Nout + col] = v;
    }
}

// ---------------- flash attention with bit-mask ----------------
// grid: H * (N/128) blocks, 256 threads (8 waves); wave = one 16-query tile
__global__ void attn_kernel(const float* __restrict__ qkv, const uint32_t* __restrict__ mask,
                            float* __restrict__ ctx) {
    __shared__ float sK[16][33];        // padded: 33 coprime with 64 banks
    __shared__ float sV[16][33];
    __shared__ float sP[8][16][17];     // per-wave P bounce buffer [query][key]

    int lane = threadIdx.x & 31;
    int wid  = threadIdx.x >> 5;
    int h    = blockIdx.x >> 5;         // 32 q-blocks per head
    int qblk = blockIdx.x & 31;
    int q0   = qblk * 128 + wid * 16;

    int mr = lane & 15;
    int kb = (lane >> 4) << 1;
    int half = lane >> 4;

    // preload Q fragments for all 8 K-steps of the 32-wide head dim
    v2f qa[8];
    {
        const float* qrow = qkv + (size_t)(q0 + mr) * 768 + h * DH;
        #pragma unroll
        for (int ds = 0; ds < 8; ++ds) {
            qa[ds].x = qrow[4 * ds + kb];
            qa[ds].y = qrow[4 * ds + kb + 1];
        }
    }

    v8f o0 = {}, o1 = {};
    float mrow[8], lrow[8];
    #pragma unroll
    for (int r = 0; r < 8; ++r) { mrow[r] = -1e30f; lrow[r] = 0.0f; }

    const float scale = 0.17677669529663688f; // 1/sqrt(32)

    for (int kt = 0; kt < NN / 16; ++kt) {
        int kbase = kt << 4;
        __syncthreads();
        // cooperative K/V tile load: 2 floats each per thread, coalesced
        #pragma unroll
        for (int i = 0; i < 2; ++i) {
            int idx = threadIdx.x + i * 256;
            int krow = idx >> 5, kcol = idx & 31;
            const float* base = qkv + (size_t)(kbase + krow) * 768 + h * DH + kcol;
            sK[krow][kcol] = base[CC];       // K at +256
            sV[krow][kcol] = base[2 * CC];   // V at +512
        }
        __syncthreads();

        // S = Q @ K^T  (8 x 16x16x4 WMMAs)
        v8f s = {};
        #pragma unroll
        for (int ds = 0; ds < 8; ++ds) {
            v2f b; b.x = sK[mr][4 * ds + kb]; b.y = sK[mr][4 * ds + kb + 1];
            s = wmma_f32(qa[ds], b, s);
        }

        // mask word for this 16-key window (one word covers it; kbase%32 in {0,16})
        uint32_t mw = 0;
        if (lane < 16) mw = mask[(size_t)(q0 + lane) * MASK_WPR + (kbase >> 5)];
        int shift = (kbase & 31) + (lane & 15);

        #pragma unroll
        for (int r = 0; r < 8; ++r) {
            int rho = r + (half << 3);
            uint32_t w = __shfl(mw, rho, 32);
            bool ok = (w >> shift) & 1u;
            float sv = s[r] * scale;
            float sm = ok ? sv : -1e30f;
            // row-max over the 16 lanes of this half-wave
            float rm = sm;
            #pragma unroll
            for (int off = 1; off < 16; off <<= 1) rm = fmaxf(rm, __shfl_xor(rm, off, 32));
            float mn = fmaxf(mrow[r], rm);
            float alpha = __expf(mrow[r] - mn);
            float p = ok ? __expf(sv - mn) : 0.0f;
            float ps = p;
            #pragma unroll
            for (int off = 1; off < 16; off <<= 1) ps += __shfl_xor(ps, off, 32);
            lrow[r] = lrow[r] * alpha + ps;
            mrow[r] = mn;
            o0[r] *= alpha;
            o1[r] *= alpha;
            sP[wid][rho][lane & 15] = p;   // C-layout -> LDS [query][key]
        }
        // LDS is in-order within a wave; fence stops compiler reordering store->load
        asm volatile("s_wait_dscnt 0" ::: "memory");

        // O += P @ V : two d-halves, each 4 x 16x16x4 WMMAs over 16 keys
        #pragma unroll
        for (int ks = 0; ks < 4; ++ks) {
            int kk = 4 * ks + kb;
            v2f pa; pa.x = sP[wid][mr][kk]; pa.y = sP[wid][mr][kk + 1];
            v2f vb0; vb0.x = sV[kk][mr];      vb0.y = sV[kk + 1][mr];
            v2f vb1; vb1.x = sV[kk][16 + mr]; vb1.y = sV[kk + 1][16 + mr];
            o0 = wmma_f32(pa, vb0, o0);
            o1 = wmma_f32(pa, vb1, o1);
        }
    }

    // normalize and write ctx[n, h*32 + d]  (self-loop guarantees lrow > 0)
    #pragma unroll
    for (int r = 0; r < 8; ++r) {
        float inv = 1.0f / lrow[r];
        int row = q0 + r + (half << 3);
        float* dst = ctx + (size_t)row * CC + h * DH;
        dst[lane & 15]        = o0[r] * inv;
        dst[16 + (lane & 15)] = o1[r] * inv;
    }
}

// ---------------- launch ----------------
extern "C" void kernel_launch(void* const* d_in, const int* in_sizes, int n_in,
                              void* d_out, int out_size, void* d_ws, size_t ws_size,
                              hipStream_t stream) {
    const float* x          = (const float*)d_in[0];
    const int*   edge_index = (const int*)d_in[1];
    const float* ln1_g      = (const float*)d_in[2];
    const float* ln1_b      = (const float*)d_in[3];
    const float* in_proj_w  = (const float*)d_in[4];
    const float* in_proj_b  = (const float*)d_in[5];
    const float* out_proj_w = (const float*)d_in[6];
    const float* out_proj_b = (const float*)d_in[7];
    const float* ln2_g      = (const float*)d_in[8];
    const float* ln2_b      = (const float*)d_in[9];
    const float* w1         = (const float*)d_in[10];
    const float* b1         = (const float*)d_in[11];
    const float* w2         = (const float*)d_in[12];
    const float* b2         = (const float*)d_in[13];
    float* out = (float*)d_out;

    int E = in_sizes[1] / 2;

    char* ws = (char*)d_ws;
    uint32_t* mask  = (uint32_t*)(ws + 0);                 // 2 MB
    float*    xnorm = (float*)(ws + (size_t)2  * 1048576); // 4 MB (reused as hnorm)
    float*    qkv   = (float*)(ws + (size_t)6  * 1048576); // 12 MB
    float*    ctx   = (float*)(ws + (size_t)18 * 1048576); // 4 MB
    float*    xres  = (float*)(ws + (size_t)22 * 1048576); // 4 MB
    float*    mid   = qkv;                                 // 16 MB (qkv+ctx region, free by then)

    const int maskWords = NN * MASK_WPR; // 524288

    zero_u32_kernel<<<(maskWords + 255) / 256, 256, 0, stream>>>(mask, maskWords);
    scatter_edges_kernel<<<(E + 255) / 256, 256, 0, stream>>>(edge_index, E, mask);

    ln_kernel<<<NN, CC, 0, stream>>>(x, ln1_g, ln1_b, xnorm);

    // QKV: [4096,768] ; tiles = 256*48 = 12288 ; 8 waves/block
    gemm_wmma_kernel<<<12288 / 8, 256, 0, stream>>>(xnorm, in_proj_w, in_proj_b, nullptr,
                                                    qkv, NN, 3 * CC, CC, 0);

    attn_kernel<<<HH * (NN / 128), 256, 0, stream>>>(qkv, mask, ctx);

    // out_proj + residual(x) -> xres ; tiles = 256*16 = 4096
    gemm_wmma_kernel<<<4096 / 8, 256, 0, stream>>>(ctx, out_proj_w, out_proj_b, x,
                                                   xres, NN, CC, CC, 1);

    ln_kernel<<<NN, CC, 0, stream>>>(xres, ln2_g, ln2_b, xnorm);

    // MLP1 + GELU -> mid ; tiles = 256*64 = 16384
    gemm_wmma_kernel<<<16384 / 8, 256, 0, stream>>>(xnorm, w1, b1, nullptr,
                                                    mid, NN, 4 * CC, CC, 2);

    // MLP2 + residual(xres) -> out ; K = 1024 ; tiles = 4096
    gemm_wmma_kernel<<<4096 / 8, 256, 0, stream>>>(mid, w2, b2, xres,
                                                   out, NN, CC, 4 * CC, 1);
}